// SAINT_85504208929395
// MI455X (gfx1250) — compile-verified
//
#include <hip/hip_runtime.h>

// ---------------------------------------------------------------------------
// GraphSAGE-style 2-layer GNN forward for MI455X (gfx1250).
// Dense GEMMs use V_WMMA_F32_16X16X4_F32 (exact f32, matches reference).
// Irregular aggregation uses L2-resident f32 atomics (agg fits in 192MB L2).
// ---------------------------------------------------------------------------

typedef float v2f __attribute__((ext_vector_type(2)));
typedef float v8f __attribute__((ext_vector_type(8)));

#define D_IN 128
#define D_H  128
#define N_CLS 64

// ---------------- utility: zero fill (graph-capture safe) -------------------
__global__ void zero_kernel(float* __restrict__ p, long long n) {
    long long i = (long long)blockIdx.x * blockDim.x + threadIdx.x;
    if (i < n) p[i] = 0.0f;
}

// ---------------- degree + inverse degree ----------------------------------
__global__ void deg_kernel(const int* __restrict__ dst, float* __restrict__ deg, int nE) {
    int e = blockIdx.x * blockDim.x + threadIdx.x;
    if (e < nE) atomicAdd(&deg[dst[e]], 1.0f);
}

__global__ void invdeg_kernel(float* __restrict__ deg, int n) {
    int i = blockIdx.x * blockDim.x + threadIdx.x;
    if (i < n) deg[i] = 1.0f / fmaxf(deg[i], 1.0f);
}

// ---------------- edge scatter: acc[dst] += X[src]  (128 floats/row) --------
// Thread t handles edge (t>>5), float4 chunk (t&31): coalesced 16B per lane.
__global__ void scatter_kernel(const float* __restrict__ X,
                               const int* __restrict__ src,
                               const int* __restrict__ dst,
                               float* __restrict__ acc, int nE) {
    long long tid = (long long)blockIdx.x * blockDim.x + threadIdx.x;
    if (tid >= (long long)nE * 32) return;
    int e = (int)(tid >> 5);
    int c = ((int)tid & 31) * 4;
    const float4 v = *(const float4*)(X + (size_t)src[e] * D_IN + c);
    float* p = acc + (size_t)dst[e] * D_IN + c;
    atomicAdd(p + 0, v.x);
    atomicAdd(p + 1, v.y);
    atomicAdd(p + 2, v.z);
    atomicAdd(p + 3, v.w);
}

// ---------------- WMMA GEMM:  Y = relu((X * inv_deg) @ W^T + b) -------------
// X: [N,128] unscaled sums; W: [128,128] row-major [out,in]; Y: [N,128].
// One wave per 16x16 output tile; K loop in steps of 4 via f32 WMMA.
__global__ __launch_bounds__(32) void gemm128_relu_kernel(
    const float* __restrict__ X, const float* __restrict__ W,
    const float* __restrict__ bias, const float* __restrict__ inv_deg,
    float* __restrict__ Y) {
    const int tile    = blockIdx.x;
    const int colTile = tile & 7;        // 128 outputs -> 8 col tiles
    const int rowTile = tile >> 3;
    const int row0 = rowTile * 16;
    const int col0 = colTile * 16;
    const int lane = threadIdx.x & 31;
    const int half = lane >> 4;          // 0: K=(k,k+1)  1: K=(k+2,k+3)
    const int l15  = lane & 15;

    // A frag: row = row0+l15 of X;  B frag: row = col0+l15 of W (i.e. W^T col)
    const float* Xrow = X + (size_t)(row0 + l15) * D_IN + 2 * half;
    const float* Wrow = W + (size_t)(col0 + l15) * D_IN + 2 * half;

    v8f acc = {};
#pragma unroll 8
    for (int k = 0; k < D_IN; k += 4) {
        v2f a = *(const v2f*)(Xrow + k);
        v2f b = *(const v2f*)(Wrow + k);
        acc = __builtin_amdgcn_wmma_f32_16x16x4_f32(
            false, a, false, b, (short)0, acc, false, false);
    }

    const float bcol = bias[col0 + l15];
#pragma unroll
    for (int r = 0; r < 8; ++r) {
        const int m = row0 + r + 8 * half;       // C/D layout: VGPR r, lane half
        float v = acc[r] * inv_deg[m] + bcol;    // fold mean after the sum
        Y[(size_t)m * D_H + col0 + l15] = fmaxf(v, 0.0f);
    }
}

// ---------------- WMMA GEMM: logits = [h1|h2] @ Wl^T + bl -------------------
// Wl: [64,256] row-major. K=256 split as 128 from h1 + 128 from h2.
__global__ __launch_bounds__(32) void gemm_logits_kernel(
    const float* __restrict__ H1, const float* __restrict__ H2,
    const float* __restrict__ Wl, const float* __restrict__ bl,
    float* __restrict__ OUT) {
    const int tile    = blockIdx.x;
    const int colTile = tile & 3;        // 64 classes -> 4 col tiles
    const int rowTile = tile >> 2;
    const int row0 = rowTile * 16;
    const int col0 = colTile * 16;
    const int lane = threadIdx.x & 31;
    const int half = lane >> 4;
    const int l15  = lane & 15;

    const float* h1row = H1 + (size_t)(row0 + l15) * D_H + 2 * half;
    const float* h2row = H2 + (size_t)(row0 + l15) * D_H + 2 * half;
    const float* wrow  = Wl + (size_t)(col0 + l15) * (2 * D_H) + 2 * half;

    v8f acc = {};
#pragma unroll 8
    for (int k = 0; k < D_H; k += 4) {
        v2f a = *(const v2f*)(h1row + k);
        v2f b = *(const v2f*)(wrow + k);
        acc = __builtin_amdgcn_wmma_f32_16x16x4_f32(
            false, a, false, b, (short)0, acc, false, false);
    }
#pragma unroll 8
    for (int k = 0; k < D_H; k += 4) {
        v2f a = *(const v2f*)(h2row + k);
        v2f b = *(const v2f*)(wrow + D_H + k);
        acc = __builtin_amdgcn_wmma_f32_16x16x4_f32(
            false, a, false, b, (short)0, acc, false, false);
    }

    const float bcol = bl[col0 + l15];
#pragma unroll
    for (int r = 0; r < 8; ++r) {
        const int m = row0 + r + 8 * half;
        OUT[(size_t)m * N_CLS + col0 + l15] = acc[r] + bcol;
    }
}

// ---------------- log_softmax over 64 classes, in place ---------------------
__global__ __launch_bounds__(64) void logsoftmax64_kernel(float* __restrict__ data) {
    __shared__ float red[N_CLS];
    const int node = blockIdx.x;
    const int c = threadIdx.x;
    float* row = data + (size_t)node * N_CLS;
    const float v = row[c];

    red[c] = v;
    __syncthreads();
#pragma unroll
    for (int s = 32; s > 0; s >>= 1) {
        if (c < s) red[c] = fmaxf(red[c], red[c + s]);
        __syncthreads();
    }
    const float mx = red[0];
    __syncthreads();

    red[c] = expf(v - mx);
    __syncthreads();
#pragma unroll
    for (int s = 32; s > 0; s >>= 1) {
        if (c < s) red[c] += red[c + s];
        __syncthreads();
    }
    const float lse = logf(red[0]) + mx;
    row[c] = v - lse;
}

// ---------------------------------------------------------------------------
extern "C" void kernel_launch(void* const* d_in, const int* in_sizes, int n_in,
                              void* d_out, int out_size, void* d_ws, size_t ws_size,
                              hipStream_t stream) {
    const float* feat = (const float*)d_in[0];
    const int*   src  = (const int*)d_in[1];
    const int*   dst  = (const int*)d_in[2];
    const float* W1   = (const float*)d_in[3];
    const float* b1   = (const float*)d_in[4];
    const float* W2   = (const float*)d_in[5];
    const float* b2   = (const float*)d_in[6];
    const float* Wl   = (const float*)d_in[7];
    const float* bl   = (const float*)d_in[8];
    float* out = (float*)d_out;

    const int N = in_sizes[0] / D_IN;     // 100000 (divisible by 16)
    const int E = in_sizes[1];            // 1600000

    // Workspace layout (floats): inv_deg[N] | agg[N*128] | h1[N*128] | h2[N*128]
    float* ws      = (float*)d_ws;
    float* inv_deg = ws;
    float* agg     = ws + N;
    float* h1      = agg + (size_t)N * D_IN;
    float* h2      = h1 + (size_t)N * D_H;

    const long long aggN = (long long)N * D_IN;
    const int rowTiles = N / 16;

    // --- degrees ---
    zero_kernel<<<(N + 255) / 256, 256, 0, stream>>>(inv_deg, N);
    deg_kernel<<<(E + 255) / 256, 256, 0, stream>>>(dst, inv_deg, E);
    invdeg_kernel<<<(N + 255) / 256, 256, 0, stream>>>(inv_deg, N);

    // --- layer 1: agg(feat) -> h1 ---
    zero_kernel<<<(int)((aggN + 255) / 256), 256, 0, stream>>>(agg, aggN);
    scatter_kernel<<<(int)(((long long)E * 32 + 255) / 256), 256, 0, stream>>>(
        feat, src, dst, agg, E);
    gemm128_relu_kernel<<<rowTiles * 8, 32, 0, stream>>>(agg, W1, b1, inv_deg, h1);

    // --- layer 2: agg(h1) -> h2 ---
    zero_kernel<<<(int)((aggN + 255) / 256), 256, 0, stream>>>(agg, aggN);
    scatter_kernel<<<(int)(((long long)E * 32 + 255) / 256), 256, 0, stream>>>(
        h1, src, dst, agg, E);
    gemm128_relu_kernel<<<rowTiles * 8, 32, 0, stream>>>(agg, W2, b2, inv_deg, h2);

    // --- head: logits -> log_softmax (in place on d_out) ---
    gemm_logits_kernel<<<rowTiles * 4, 32, 0, stream>>>(h1, h2, Wl, bl, out);
    logsoftmax64_kernel<<<N, 64, 0, stream>>>(out);
}